// _GTSEdgeTransformerNodeModel_62002147885264
// MI455X (gfx1250) — compile-verified
//
#include <hip/hip_runtime.h>
#include <hip/hip_bf16.h>
#include <math.h>

typedef __attribute__((ext_vector_type(2))) float v2f;
typedef __attribute__((ext_vector_type(8))) float v8f;

#define B_   8
#define NN_  40
#define H_   128
#define NH_  8
#define DK_  16
#define L_   3
#define TOK_ (B_*NN_*NN_)   /* 12800 edge tokens */
#define NND_ (B_*NN_)       /* 320 nodes */
#define EDG_ 3840
#define BN_SCALE_ 0.9999950000374997f   /* 1/sqrt(1+1e-5) */

// ---------------------------------------------------------------------------
// fp32 GEMM via V_WMMA_F32_16X16X4_F32.
// Each wave computes a 64x64 block of C = act(A[MxK] @ W[KxN] + bias):
// 4 M-tiles x 4 N-tiles of 16x16, so each B fragment feeds 4 WMMAs.
// Per K=4 slab: 4x b64 (A) + 8x b32 (B) loads, 16 WMMAs.
// Block = 128 threads = 4 waves (256 rows). grid.y = N/64.
// Requires M % 64 == 0 (true for 320 / 3840 / 12800) and N % 64 == 0.
// ---------------------------------------------------------------------------
__global__ void __launch_bounds__(128) gemm_wmma_f32(
    const float* __restrict__ A, const float* __restrict__ W,
    const float* __restrict__ bias, float* __restrict__ C,
    int M, int K, int N, int relu_act)
{
  const int lane = threadIdx.x & 31;
  const int wid  = (blockIdx.x * (int)blockDim.x + (int)threadIdx.x) >> 5;
  const int m0   = wid * 64;
  if (m0 >= M) return;                     // wave-uniform: EXEC stays all-1s
  const int n0   = blockIdx.y * 64;
  const int mrow = lane & 15;              // row within a 16-tile
  const int kh   = lane >> 4;              // which K pair this lane-half holds

  const float* Arow[4];
#pragma unroll
  for (int mt = 0; mt < 4; ++mt)
    Arow[mt] = A + (size_t)(m0 + 16 * mt + mrow) * K + 2 * kh;

  v8f acc[4][4] = {};
  for (int k0 = 0; k0 < K; k0 += 4) {
    // A fragments: lane holds A[row, k0 + 2*kh + {0,1}] (contiguous pair)
    v2f a[4];
#pragma unroll
    for (int mt = 0; mt < 4; ++mt) a[mt] = *(const v2f*)(Arow[mt] + k0);
    const float* Wr = W + (size_t)(k0 + 2 * kh) * N + n0 + mrow;
#pragma unroll
    for (int t = 0; t < 4; ++t) {
      // B fragment: lane holds W[k0 + 2*kh + {0,1}, col]
      v2f b;
      b.x = Wr[t * 16];
      b.y = Wr[(size_t)N + t * 16];
#pragma unroll
      for (int mt = 0; mt < 4; ++mt)
        acc[mt][t] = __builtin_amdgcn_wmma_f32_16x16x4_f32(
            false, a[mt], false, b, (short)0, acc[mt][t], false, false);
    }
  }

#pragma unroll
  for (int t = 0; t < 4; ++t) {
    const int col = n0 + t * 16 + mrow;
    const float bb = bias ? bias[col] : 0.0f;
#pragma unroll
    for (int mt = 0; mt < 4; ++mt) {
#pragma unroll
      for (int r = 0; r < 8; ++r) {
        const int row = m0 + 16 * mt + r + 8 * kh;  // C layout: VGPR r -> rows r / r+8
        float v = acc[mt][t][r] + bb;
        if (relu_act) v = fmaxf(v, 0.0f);
        C[(size_t)row * N + col] = v;
      }
    }
  }
}

// ---------------------------------------------------------------------------
// LayerNorm over H=128, one wave per token, optional fused residual add.
// ---------------------------------------------------------------------------
__global__ void __launch_bounds__(256) ln_kernel(
    const float* __restrict__ a, const float* __restrict__ add,
    const float* __restrict__ g, const float* __restrict__ be,
    float* __restrict__ out, int T)
{
  const int w    = (blockIdx.x * (int)blockDim.x + (int)threadIdx.x) >> 5;
  const int lane = threadIdx.x & 31;
  if (w >= T) return;
  const size_t base = (size_t)w * H_;
  float v[4];
#pragma unroll
  for (int i = 0; i < 4; ++i) {
    const int c = lane + 32 * i;
    v[i] = a[base + c];
    if (add) v[i] += add[base + c];
  }
  float s = v[0] + v[1] + v[2] + v[3];
#pragma unroll
  for (int m = 16; m >= 1; m >>= 1) s += __shfl_xor(s, m, 32);
  const float mean = s * (1.0f / 128.0f);
  float q = 0.0f;
#pragma unroll
  for (int i = 0; i < 4; ++i) { const float d = v[i] - mean; q += d * d; }
#pragma unroll
  for (int m = 16; m >= 1; m >>= 1) q += __shfl_xor(q, m, 32);
  const float rs = rsqrtf(q * (1.0f / 128.0f) + 1e-5f);
#pragma unroll
  for (int i = 0; i < 4; ++i) {
    const int c = lane + 32 * i;
    out[base + c] = (v[i] - mean) * rs * g[c] + be[c];
  }
}

// ---------------------------------------------------------------------------
// Composer init: xe[b,i,j,c] = pl[b,i,c] + pr[b,j,c] + comp_b[c]
// ---------------------------------------------------------------------------
__global__ void compose_init(const float* __restrict__ pl,
                             const float* __restrict__ pr,
                             const float* __restrict__ cb,
                             float* __restrict__ xe)
{
  const int idx = blockIdx.x * 256 + threadIdx.x;     // TOK_*H_ total, exact
  const int c = idx & (H_ - 1);
  int t = idx >> 7;
  const int j = t % NN_; t /= NN_;
  const int i = t % NN_;
  const int b = t / NN_;
  xe[idx] = pl[(size_t)(b * NN_ + i) * H_ + c] +
            pr[(size_t)(b * NN_ + j) * H_ + c] + cb[c];
}

// ---------------------------------------------------------------------------
// Edge scatter-add: edense accumulation into xe
// ---------------------------------------------------------------------------
__global__ void scatter_edges(const float* __restrict__ ea,
                              const int* __restrict__ eidx,
                              float* __restrict__ xe)
{
  const int idx = blockIdx.x * 256 + threadIdx.x;     // EDG_*H_ total, exact
  const int c = idx & (H_ - 1);
  const int e = idx >> 7;
  const int s0 = eidx[e];
  const int s1 = eidx[EDG_ + e];
  const int bi = s0 / NN_, ii = s0 % NN_, jj = s1 % NN_;
  atomicAdd(&xe[((size_t)(bi * NN_ + ii) * NN_ + jj) * H_ + c], ea[(size_t)e * H_ + c]);
}

// ---------------------------------------------------------------------------
// Triangular attention: thread = (b,x,y,h); softmax over shared node a.
// scores staged in LDS (128 threads * 40 floats = 20 KB).
// ---------------------------------------------------------------------------
__global__ void __launch_bounds__(128) attn_kernel(
    const float* __restrict__ lk, const float* __restrict__ rk,
    const float* __restrict__ lv, const float* __restrict__ rv,
    float* __restrict__ o)
{
  __shared__ float smem[128 * NN_];
  const int tid = blockIdx.x * 128 + threadIdx.x;     // B*NN*NN*NH total, exact
  const int h = tid & (NH_ - 1);
  int t = tid >> 3;
  const int y = t % NN_; t /= NN_;
  const int x = t % NN_;
  const int b = t / NN_;
  float* s = smem + (int)threadIdx.x * NN_;
  const float scale = 0.25f;                          // 1/sqrt(DK=16)

  const size_t lbase = (size_t)(b * NN_ + x) * NN_ * H_ + h * DK_;  // + a*H_
  const size_t rbase = (size_t)b * NN_ * NN_ * H_ + (size_t)y * H_ + h * DK_; // + a*NN_*H_

  float mx = -1e30f;
  for (int a = 0; a < NN_; ++a) {
    const float* lp = lk + lbase + (size_t)a * H_;
    const float* rp = rk + rbase + (size_t)a * NN_ * H_;
    float acc = 0.0f;
#pragma unroll
    for (int d = 0; d < DK_; ++d) acc += lp[d] * rp[d];
    acc *= scale;
    s[a] = acc;
    mx = fmaxf(mx, acc);
  }
  float denom = 0.0f;
  for (int a = 0; a < NN_; ++a) { const float p = expf(s[a] - mx); s[a] = p; denom += p; }
  const float inv = 1.0f / denom;

  float od[DK_] = {};
  for (int a = 0; a < NN_; ++a) {
    const float p = s[a] * inv;
    const float* lvp = lv + lbase + (size_t)a * H_;
    const float* rvp = rv + rbase + (size_t)a * NN_ * H_;
#pragma unroll
    for (int d = 0; d < DK_; ++d) od[d] += p * lvp[d] * rvp[d];
  }
  float* op = o + ((size_t)(b * NN_ + x) * NN_ + y) * H_ + h * DK_;
#pragma unroll
  for (int d = 0; d < DK_; ++d) op[d] = od[d];
}

// ---------------------------------------------------------------------------
// Eval-mode BatchNorm (running stats = 0/1): y = x*(BN_SCALE*g[c]) + b[c]
// ---------------------------------------------------------------------------
__global__ void bn_kernel(const float* __restrict__ in,
                          const float* __restrict__ g, const float* __restrict__ b,
                          float* __restrict__ out, int total, int C)
{
  const int idx = blockIdx.x * 256 + threadIdx.x;
  if (idx >= total) return;
  const int c = idx % C;
  out[idx] = in[idx] * (BN_SCALE_ * g[c]) + b[c];
}

// ---------------------------------------------------------------------------
// Decomposer row/col sums + fused node BatchNorm:
// nf[b,i,c] = BN( sum_j deco[b,i,j,c] + sum_j deco[b,j,i,H+c] )
// ---------------------------------------------------------------------------
__global__ void rowcol_kernel(const float* __restrict__ deco,
                              const float* __restrict__ g, const float* __restrict__ be,
                              float* __restrict__ nf)
{
  const int idx = blockIdx.x * 256 + threadIdx.x;     // NND_*H_ total, exact
  const int c = idx & (H_ - 1);
  const int i = (idx >> 7) % NN_;
  const int b = (idx >> 7) / NN_;
  float s = 0.0f;
  for (int j = 0; j < NN_; ++j) {
    s += deco[((size_t)(b * NN_ + i) * NN_ + j) * (2 * H_) + c];
    s += deco[((size_t)(b * NN_ + j) * NN_ + i) * (2 * H_) + H_ + c];
  }
  nf[idx] = s * (BN_SCALE_ * g[c]) + be[c];
}

// ---------------------------------------------------------------------------
// Fused head MLP: 128 -> relu 64 -> relu 32 -> 1, one block per node.
// ---------------------------------------------------------------------------
__global__ void __launch_bounds__(64) head_kernel(
    const float* __restrict__ nf,
    const float* __restrict__ w1, const float* __restrict__ b1,
    const float* __restrict__ w2, const float* __restrict__ b2,
    const float* __restrict__ w3, const float* __restrict__ b3,
    float* __restrict__ out)
{
  __shared__ float h1[64];
  __shared__ float h2[32];
  const int node = blockIdx.x;
  const int t = threadIdx.x;
  const float* xp = nf + (size_t)node * H_;
  float s = b1[t];
  for (int k = 0; k < H_; ++k) s += xp[k] * w1[k * 64 + t];
  h1[t] = fmaxf(s, 0.0f);
  __syncthreads();
  if (t < 32) {
    float u = b2[t];
    for (int k = 0; k < 64; ++k) u += h1[k] * w2[k * 32 + t];
    h2[t] = fmaxf(u, 0.0f);
  }
  __syncthreads();
  if (t == 0) {
    float u = b3[0];
    for (int k = 0; k < 32; ++k) u += h2[k] * w3[k];
    out[node] = u;
  }
}

// ---------------------------------------------------------------------------
extern "C" void kernel_launch(void* const* d_in, const int* in_sizes, int n_in,
                              void* d_out, int out_size, void* d_ws, size_t ws_size,
                              hipStream_t stream) {
  (void)in_sizes; (void)n_in; (void)out_size; (void)ws_size;

  const float* x          = (const float*)d_in[0];
  const float* edge_attr  = (const float*)d_in[1];
  const float* node_enc_w = (const float*)d_in[2];
  const float* node_enc_b = (const float*)d_in[3];
  const float* edge_enc_w = (const float*)d_in[4];
  const float* edge_enc_b = (const float*)d_in[5];
  const float* comp_w     = (const float*)d_in[6];
  const float* comp_b     = (const float*)d_in[7];
  const float* ln1_g      = (const float*)d_in[8];
  const float* ln1_b      = (const float*)d_in[9];
  const float* w_lk       = (const float*)d_in[10];
  const float* w_rk       = (const float*)d_in[11];
  const float* w_lv       = (const float*)d_in[12];
  const float* w_rv       = (const float*)d_in[13];
  const float* w_out      = (const float*)d_in[14];
  const float* ffn_w1     = (const float*)d_in[15];
  const float* ffn_b1     = (const float*)d_in[16];
  const float* ffn_w2     = (const float*)d_in[17];
  const float* ffn_b2     = (const float*)d_in[18];
  const float* ln_agg_g   = (const float*)d_in[19];
  const float* ln_agg_b   = (const float*)d_in[20];
  const float* ln2_g      = (const float*)d_in[21];
  const float* ln2_b      = (const float*)d_in[22];
  const float* dec_bn_g   = (const float*)d_in[23];
  const float* dec_bn_b   = (const float*)d_in[24];
  const float* dec_w1     = (const float*)d_in[25];
  const float* dec_b1     = (const float*)d_in[26];
  const float* dec_w2     = (const float*)d_in[27];
  const float* dec_b2     = (const float*)d_in[28];
  const float* nm_bn_g    = (const float*)d_in[29];
  const float* nm_bn_b    = (const float*)d_in[30];
  const float* nm_w1      = (const float*)d_in[31];
  const float* nm_b1      = (const float*)d_in[32];
  const float* nm_w2      = (const float*)d_in[33];
  const float* nm_b2      = (const float*)d_in[34];
  const float* head_w1    = (const float*)d_in[35];
  const float* head_b1    = (const float*)d_in[36];
  const float* head_w2    = (const float*)d_in[37];
  const float* head_b2    = (const float*)d_in[38];
  const float* head_w3    = (const float*)d_in[39];
  const float* head_b3    = (const float*)d_in[40];
  const int*   eidx       = (const int*)d_in[41];
  float* out = (float*)d_out;

  // ---- workspace layout (floats) ----
  float* ws = (float*)d_ws;
  size_t off = 0;
  auto alloc = [&](size_t n) { float* p = ws + off; off += n; return p; };
  float* xn   = alloc((size_t)NND_ * H_);
  float* pl   = alloc((size_t)NND_ * H_);
  float* pr   = alloc((size_t)NND_ * H_);
  float* ea   = alloc((size_t)EDG_ * H_);
  float* xe   = alloc((size_t)TOK_ * H_);
  float* z    = alloc((size_t)TOK_ * H_);   // reused: z / o_attn / dec bn input
  float* k1   = alloc((size_t)TOK_ * H_);   // reused: lk / o / y
  float* k2   = alloc((size_t)TOK_ * H_);   // reused: rk / xf
  float* v1   = alloc((size_t)TOK_ * H_);   // lv
  float* v2   = alloc((size_t)TOK_ * H_);   // rv
  float* hb   = alloc((size_t)TOK_ * 2 * H_); // FFN / dec hidden
  float* hb2  = alloc((size_t)TOK_ * 2 * H_); // dec output (2H)
  float* nf   = alloc((size_t)NND_ * H_);
  float* nf2  = alloc((size_t)NND_ * H_);
  float* nf3  = alloc((size_t)NND_ * H_);

  auto gemm = [&](const float* A, const float* W, const float* bias, float* C,
                  int M, int K, int N, int act) {
    // one wave per 64x64 block; 4 waves per workgroup
    dim3 grid((M / 64 + 3) / 4, N / 64);
    gemm_wmma_f32<<<grid, 128, 0, stream>>>(A, W, bias, C, M, K, N, act);
  };
  auto ln = [&](const float* a, const float* add, const float* g, const float* b,
                float* o, int T) {
    ln_kernel<<<(T * 32 + 255) / 256, 256, 0, stream>>>(a, add, g, b, o, T);
  };

  // ---- encoders + composer ----
  gemm(x,         node_enc_w, node_enc_b, xn, NND_, 64, H_, 0);
  gemm(edge_attr, edge_enc_w, edge_enc_b, ea, EDG_, 16, H_, 0);
  gemm(xn, comp_w,            nullptr, pl, NND_, H_, H_, 0);   // top half of comp_w
  gemm(xn, comp_w + H_ * H_,  nullptr, pr, NND_, H_, H_, 0);   // bottom half
  compose_init<<<(TOK_ * H_) / 256, 256, 0, stream>>>(pl, pr, comp_b, xe);
  scatter_edges<<<(EDG_ * H_) / 256, 256, 0, stream>>>(ea, eidx, xe);

  // ---- transformer layers ----
  for (int l = 0; l < L_; ++l) {
    const size_t oH = (size_t)l * H_, oHH = (size_t)l * H_ * H_;
    const size_t oHF = (size_t)l * H_ * 2 * H_, oF = (size_t)l * 2 * H_;
    ln(xe, nullptr, ln1_g + oH, ln1_b + oH, z, TOK_);
    gemm(z, w_lk + oHH, nullptr, k1, TOK_, H_, H_, 0);
    gemm(z, w_rk + oHH, nullptr, k2, TOK_, H_, H_, 0);
    gemm(z, w_lv + oHH, nullptr, v1, TOK_, H_, H_, 0);
    gemm(z, w_rv + oHH, nullptr, v2, TOK_, H_, H_, 0);
    attn_kernel<<<(B_ * NN_ * NN_ * NH_) / 128, 128, 0, stream>>>(k1, k2, v1, v2, z);
    gemm(z, w_out + oHH, nullptr, k1, TOK_, H_, H_, 0);        // o
    ln(xe, k1, ln_agg_g + oH, ln_agg_b + oH, k2, TOK_);        // xf = LN(xe + o)
    gemm(k2, ffn_w1 + oHF, ffn_b1 + oF, hb, TOK_, H_, 2 * H_, 1);
    gemm(hb, ffn_w2 + oHF, ffn_b2 + oH, k1, TOK_, 2 * H_, H_, 0); // y
    ln(k2, k1, ln2_g + oH, ln2_b + oH, xe, TOK_);              // xe = LN(xf + y)
  }

  // ---- decomposer + node MLP + head ----
  bn_kernel<<<(TOK_ * H_) / 256, 256, 0, stream>>>(xe, dec_bn_g, dec_bn_b, z, TOK_ * H_, H_);
  gemm(z,  dec_w1, dec_b1, hb,  TOK_, H_,     2 * H_, 1);
  gemm(hb, dec_w2, dec_b2, hb2, TOK_, 2 * H_, 2 * H_, 0);
  rowcol_kernel<<<(NND_ * H_) / 256, 256, 0, stream>>>(hb2, nm_bn_g, nm_bn_b, nf);
  gemm(nf,  nm_w1, nm_b1, nf2, NND_, H_, H_, 1);
  gemm(nf2, nm_w2, nm_b2, nf3, NND_, H_, H_, 0);
  head_kernel<<<NND_, 64, 0, stream>>>(nf3, head_w1, head_b1, head_w2, head_b2,
                                       head_w3, head_b3, out);
}